// MoEGate_36971078484477
// MI455X (gfx1250) — compile-verified
//
#include <hip/hip_runtime.h>
#include <hip/hip_bf16.h>

typedef float v2f __attribute__((ext_vector_type(2)));
typedef float v8f __attribute__((ext_vector_type(8)));

#define MG_DIM   2048
#define MG_NEXP  64
#define MG_NTOK  16384   // BSZ * SEQ
#define MG_SEQ   4096
#define MG_BSZ   4
#define MG_TOPK  2
#define MG_ALPHA 0.01f

// ---------------------------------------------------------------------------
// Main kernel: router GEMM (fp32 WMMA 16x16x4) + softmax + top-2 + aux partials
// Grid: MG_NTOK/64 = 256 blocks, 128 threads (4 waves). Wave w handles tokens
// [blk*64 + w*16, +16) x all 64 experts via four 16x16 f32 accumulators.
// ---------------------------------------------------------------------------
__global__ __launch_bounds__(128) void moe_gate_main(
    const float* __restrict__ X,       // [MG_NTOK, MG_DIM]
    const float* __restrict__ W,       // [MG_NEXP, MG_DIM]
    float* __restrict__ out_idx,       // [MG_NTOK, 2] (indices stored as float)
    float* __restrict__ out_w,         // [MG_NTOK, 2]
    float* __restrict__ ws_score,      // [MG_BSZ * MG_NEXP] global accum
    float* __restrict__ ws_cnt)        // [MG_BSZ * MG_NEXP] global accum
{
    __shared__ float lds_logits[64 * MG_NEXP];  // 64 tokens x 64 experts = 16 KB
    __shared__ float lds_score[MG_NEXP];
    __shared__ float lds_cnt[MG_NEXP];

    const int tid  = threadIdx.x;
    const int wave = tid >> 5;
    const int lane = tid & 31;
    const int half = lane >> 4;    // 0: lanes 0-15, 1: lanes 16-31
    const int l16  = lane & 15;

    if (tid < MG_NEXP) { lds_score[tid] = 0.0f; lds_cnt[tid] = 0.0f; }

    const int tok0 = blockIdx.x * 64 + wave * 16;

    // A fragment (16x4 f32): VGPR0 = K=0 (lanes 0-15) / K=2 (lanes 16-31),
    // VGPR1 = K=1 / K=3, row M = lane&15.  -> per lane one b64 load per step.
    const float* pA  = X + (size_t)(tok0 + l16) * MG_DIM + (half << 1);
    // B fragment (4x16 f32), mirrored layout: column N = lane&15, K split by half.
    const float* pB0 = W + (size_t)(l16) * MG_DIM + (half << 1);
    const float* pB1 = pB0 + (size_t)16 * MG_DIM;
    const float* pB2 = pB0 + (size_t)32 * MG_DIM;
    const float* pB3 = pB0 + (size_t)48 * MG_DIM;

    v8f acc0 = {}; v8f acc1 = {}; v8f acc2 = {}; v8f acc3 = {};

#pragma unroll 4
    for (int k = 0; k < MG_DIM; k += 4) {
        v2f a  = *(const v2f*)(pA  + k);
        v2f b0 = *(const v2f*)(pB0 + k);
        v2f b1 = *(const v2f*)(pB1 + k);
        v2f b2 = *(const v2f*)(pB2 + k);
        v2f b3 = *(const v2f*)(pB3 + k);
        // 8-arg form: (neg_a, A, neg_b, B, c_mod, C, reuse_a, reuse_b)
        acc0 = __builtin_amdgcn_wmma_f32_16x16x4_f32(false, a, false, b0, (short)0, acc0, false, false);
        acc1 = __builtin_amdgcn_wmma_f32_16x16x4_f32(false, a, false, b1, (short)0, acc1, false, false);
        acc2 = __builtin_amdgcn_wmma_f32_16x16x4_f32(false, a, false, b2, (short)0, acc2, false, false);
        acc3 = __builtin_amdgcn_wmma_f32_16x16x4_f32(false, a, false, b3, (short)0, acc3, false, false);
    }

    // C/D layout: VGPR r holds row M = r + 8*half, column N = lane&15.
    const int trow = wave * 16 + 8 * half;
#pragma unroll
    for (int r = 0; r < 8; ++r) {
        float* dst = &lds_logits[(size_t)(trow + r) * MG_NEXP + l16];
        dst[0]  = acc0[r];
        dst[16] = acc1[r];
        dst[32] = acc2[r];
        dst[48] = acc3[r];
    }
    __syncthreads();

    // ---- per-token softmax + top-2 (threads 0..63, one token each) ----
    if (tid < 64) {
        const int n = blockIdx.x * 64 + tid;          // global token id
        const float* row = &lds_logits[(size_t)tid * MG_NEXP];

        float l1 = -3.402823466e38f, l2 = -3.402823466e38f;
        int   i1 = 0, i2 = 0;
#pragma unroll 8
        for (int e = 0; e < MG_NEXP; ++e) {
            float v = row[e];
            if (v > l1)      { l2 = l1; i2 = i1; l1 = v; i1 = e; }
            else if (v > l2) { l2 = v;  i2 = e; }
        }
        const float m = l1;
        float s = 0.0f;
#pragma unroll 8
        for (int e = 0; e < MG_NEXP; ++e) s += __expf(row[e] - m);
        const float inv = 1.0f / s;

        // accumulate softmax scores per expert (staggered to spread LDS banks)
        for (int e = 0; e < MG_NEXP; ++e) {
            int j = (e + tid) & (MG_NEXP - 1);
            atomicAdd(&lds_score[j], __expf(row[j] - m) * inv);
        }

        const float s1 = inv;                       // exp(0)/s
        const float s2 = __expf(l2 - m) * inv;
        const float wnorm = 1.0f / (s1 + s2 + 1e-20f);

        out_idx[2 * n + 0] = (float)i1;
        out_idx[2 * n + 1] = (float)i2;
        out_w[2 * n + 0]   = s1 * wnorm;
        out_w[2 * n + 1]   = s2 * wnorm;

        atomicAdd(&lds_cnt[i1], 1.0f);
        atomicAdd(&lds_cnt[i2], 1.0f);
    }
    __syncthreads();

    // one batch per block (64 tokens all inside one 4096-token batch)
    if (tid < MG_NEXP) {
        const int b = (blockIdx.x * 64) >> 12;        // /4096
        atomicAdd(&ws_score[b * MG_NEXP + tid], lds_score[tid]);
        atomicAdd(&ws_cnt  [b * MG_NEXP + tid], lds_cnt[tid]);
    }
}

// ---------------------------------------------------------------------------
// Finalize: aux = ALPHA * E/(B*S*S*K) * sum_{b,e} cnt[b,e]*sumscore[b,e]
// ---------------------------------------------------------------------------
__global__ __launch_bounds__(256) void moe_gate_finalize(
    const float* __restrict__ ws_score,
    const float* __restrict__ ws_cnt,
    float* __restrict__ out_aux)
{
    __shared__ float red[256];
    const int t = threadIdx.x;
    red[t] = ws_cnt[t] * ws_score[t];
    __syncthreads();
#pragma unroll
    for (int s = 128; s > 0; s >>= 1) {
        if (t < s) red[t] += red[t + s];
        __syncthreads();
    }
    if (t == 0) {
        const float scale = (float)((double)MG_ALPHA * (double)MG_NEXP /
            ((double)MG_BSZ * (double)MG_SEQ * (double)MG_SEQ * (double)MG_TOPK));
        out_aux[0] = red[0] * scale;
    }
}

// ---------------------------------------------------------------------------
extern "C" void kernel_launch(void* const* d_in, const int* in_sizes, int n_in,
                              void* d_out, int out_size, void* d_ws, size_t ws_size,
                              hipStream_t stream) {
    const float* X = (const float*)d_in[0];   // hidden_states [4,4096,2048] fp32
    const float* W = (const float*)d_in[1];   // weight [64,2048] fp32
    float* out      = (float*)d_out;          // [idx(32768) | weight(32768) | aux(1)]
    float* ws_score = (float*)d_ws;                       // 256 floats
    float* ws_cnt   = ws_score + MG_BSZ * MG_NEXP;        // 256 floats

    hipMemsetAsync(d_ws, 0, 2 * MG_BSZ * MG_NEXP * sizeof(float), stream);

    moe_gate_main<<<MG_NTOK / 64, 128, 0, stream>>>(
        X, W, out, out + 2 * MG_NTOK, ws_score, ws_cnt);

    moe_gate_finalize<<<1, 256, 0, stream>>>(ws_score, ws_cnt, out + 4 * MG_NTOK);
}